// AdaptiveLayer_16252156248548
// MI455X (gfx1250) — compile-verified
//
#include <hip/hip_runtime.h>

// AdaptiveLayer scan: out[b,t,f] = max(x[b,t,f] - a, 0); a = (a + 0.1*out)*0.9
// x: [B=32, T=512, F=4096] f32. Serial in T, parallel over B*F lanes.
// HBM-bound (512 MB traffic, ~22 us floor at 23.3 TB/s). Strategy: 16-deep
// async global->LDS ring buffer per thread (ASYNCcnt pipelining), adapt state
// in registers, nontemporal streaming stores.

#define ADAPT_RATE    0.1f
#define RECOVERY_RATE 0.1f

constexpr int kB = 32;
constexpr int kT = 512;
constexpr int kF = 4096;

constexpr int VEC    = 4;            // float4 per thread
constexpr int TPB    = 256;          // threads per block (8 wave32)
constexpr int FCHUNK = TPB * VEC;    // 1024 floats of one row per block
constexpr int STAGES = 16;           // async pipeline depth (ring buffer)
// LDS: STAGES * FCHUNK * 4 B = 64 KB per workgroup (<< 320 KB/WGP)

typedef float float4_t __attribute__((ext_vector_type(4)));
typedef int   v4i      __attribute__((ext_vector_type(4)));
typedef __attribute__((address_space(1))) v4i gv4i;   // global int4
typedef __attribute__((address_space(3))) v4i lv4i;   // LDS int4

__global__ __launch_bounds__(TPB) void adaptive_scan_async_kernel(
    const float* __restrict__ x,        // [B, T, F]
    const float* __restrict__ adapt0,   // [1, F] (broadcast over B)
    float* __restrict__ out)            // [B, T, F]
{
    __shared__ float lds_ring[STAGES * FCHUNK];   // 64 KB ring buffer

    const int tid = threadIdx.x;
    const int blk = blockIdx.x;                   // 0..127
    const int b    = blk / (kF / FCHUNK);         // 4 blocks per batch row
    const int foff = (blk % (kF / FCHUNK)) * FCHUNK + tid * VEC;

    const size_t base = (size_t)b * kT * kF + foff;
    const float* gsrc = x + base;                 // column base, stride F per t
    float*       gdst = out + base;

    // Each lane owns its private 16 B slot in every stage: no barriers needed.
    float* lds_lane = &lds_ring[tid * VEC];

    // Initial adaptation state (broadcast of [1,F] over B), carried in VGPRs.
    float4_t a = *(const float4_t*)(adapt0 + foff);

    // ---- Prologue: fill the async pipeline (ASYNCcnt = STAGES <= 63) ----
#pragma unroll
    for (int s = 0; s < STAGES; ++s) {
        __builtin_amdgcn_global_load_async_to_lds_b128(
            (gv4i*)(gsrc + (size_t)s * kF),
            (lv4i*)(lds_lane + s * FCHUNK),
            /*imm offset*/ 0, /*cpol*/ 0);
    }

    // ---- Steady state: wait oldest, consume, compute, store, refill ----
    for (int t = 0; t < kT; ++t) {
        // Async loads complete in order: <= STAGES-1 outstanding means the
        // load for timestep t has landed in LDS.
        __builtin_amdgcn_s_wait_asynccnt(STAGES - 1);

        const int slot = t & (STAGES - 1);
        float4_t v = *(const float4_t*)(lds_lane + slot * FCHUNK);

        float4_t o = v - a;
        o.x = fmaxf(o.x, 0.0f);
        o.y = fmaxf(o.y, 0.0f);
        o.z = fmaxf(o.z, 0.0f);
        o.w = fmaxf(o.w, 0.0f);
        // a = (a + 0.1*o) * 0.9   (matches reference op order)
        float4_t an = a;
        an.x = (a.x + ADAPT_RATE * o.x) * (1.0f - RECOVERY_RATE);
        an.y = (a.y + ADAPT_RATE * o.y) * (1.0f - RECOVERY_RATE);
        an.z = (a.z + ADAPT_RATE * o.z) * (1.0f - RECOVERY_RATE);
        an.w = (a.w + ADAPT_RATE * o.w) * (1.0f - RECOVERY_RATE);
        a = an;

        // Write-once output: nontemporal streaming store.
        __builtin_nontemporal_store(o, (float4_t*)(gdst + (size_t)t * kF));

        // Unconditionally refill the just-consumed slot to keep exactly
        // STAGES loads outstanding (tail iterations clamp to the last row;
        // those slots are never read again, so the redundant fill is benign).
        int tn = t + STAGES;
        if (tn >= kT) tn = kT - 1;
        __builtin_amdgcn_global_load_async_to_lds_b128(
            (gv4i*)(gsrc + (size_t)tn * kF),
            (lv4i*)(lds_lane + slot * FCHUNK),
            0, 0);
    }
    // S_ENDPGM performs an implicit wait-idle for the remaining async ops.
}

extern "C" void kernel_launch(void* const* d_in, const int* in_sizes, int n_in,
                              void* d_out, int out_size, void* d_ws, size_t ws_size,
                              hipStream_t stream) {
    const float* x          = (const float*)d_in[0];   // [32, 512, 4096] f32
    const float* adaptation = (const float*)d_in[1];   // [1, 4096] f32
    float* out              = (float*)d_out;           // [32, 512, 4096] f32

    const int grid = kB * (kF / FCHUNK);               // 32 * 4 = 128 blocks
    adaptive_scan_async_kernel<<<grid, TPB, 0, stream>>>(x, adaptation, out);
}